// Wide_51608327029121
// MI455X (gfx1250) — compile-verified
//
#include <hip/hip_runtime.h>

// ---------------------------------------------------------------------------
// Wide & sparse linear: out[b] = dense[b,:13] . W[:13] + sum_i W[off_i + s_i[b]]
// BATCH = 16384, IN_FEATURES = 19,600,013.
// Memory/latency-bound gather kernel; dense matvec on the WMMA pipe
// (V_WMMA_F32_16X16X4_F32, two 16-row tiles per wave32, branch-free K padding).
// ---------------------------------------------------------------------------

typedef __attribute__((ext_vector_type(2))) float v2f;
typedef __attribute__((ext_vector_type(8))) float v8f;

#define WIDE_BATCH     16384
#define WIDE_NDENSE    13
#define WIDE_BLOCK     128
#define WIDE_DENSE_LAST (WIDE_BATCH * WIDE_NDENSE - 1)   // 212,991

// cumulative vocab offsets into W (after the 13 dense weights)
#define OFF0 13
#define OFF1 1000013
#define OFF2 2000013
#define OFF3 7000013
#define OFF4 9000013
#define OFF5 9500013
#define OFF6 9600013
#define OFF7 14600013

#if defined(__AMDGCN__) && __has_builtin(__builtin_amdgcn_wmma_f32_16x16x4_f32)
#define WIDE_HAVE_WMMA 1
#endif

__device__ __forceinline__ float sel8(v8f v, int j) {
  // 7 v_cndmask chain; avoids scratch from dynamic vector subscript
  float r = v[0];
  r = (j == 1) ? v[1] : r;
  r = (j == 2) ? v[2] : r;
  r = (j == 3) ? v[3] : r;
  r = (j == 4) ? v[4] : r;
  r = (j == 5) ? v[5] : r;
  r = (j == 6) ? v[6] : r;
  r = (j == 7) ? v[7] : r;
  return r;
}

__global__ __launch_bounds__(WIDE_BLOCK)
void wide_gather_wmma_kernel(const float* __restrict__ dense,
                             const float* __restrict__ W,
                             const int* __restrict__ s0, const int* __restrict__ s1,
                             const int* __restrict__ s2, const int* __restrict__ s3,
                             const int* __restrict__ s4, const int* __restrict__ s5,
                             const int* __restrict__ s6, const int* __restrict__ s7,
                             float* __restrict__ out)
{
  const int tid  = blockIdx.x * WIDE_BLOCK + threadIdx.x;  // one row per lane
  const int lane = threadIdx.x & 31;
  const int base = tid & ~31;                              // first row of this wave's tile

  // ---- sparse part: 8 irregular gathers per row, one row per lane ---------
  // all 8 loads issued before the accumulate -> latency overlapped
  float sacc;
  sacc  = W[OFF0 + s0[tid]];
  sacc += W[OFF1 + s1[tid]];
  sacc += W[OFF2 + s2[tid]];
  sacc += W[OFF3 + s3[tid]];
  sacc += W[OFF4 + s4[tid]];
  sacc += W[OFF5 + s5[tid]];
  sacc += W[OFF6 + s6[tid]];
  sacc += W[OFF7 + s7[tid]];

  float dval;
#if defined(WIDE_HAVE_WMMA)
  // ---- dense part on the matrix pipe: D(16x16) = A(16x4) x B(4x16) + C ----
  // A layout (ISA 7.12.2, 32-bit A 16x4): lanes 0-15 hold K={0,1} in v[0],v[1];
  //                                       lanes 16-31 hold K={2,3}.
  // B 4x16 (row striped across lanes, C/D convention): VGPR v, half h -> K=v+2h.
  // B replicates W[k] across all 16 columns -> every column of D is the dot.
  // K padding 13->16 is branch-free: only B is zeroed for k>=13 (value select,
  // lowers to v_cndmask); A addresses are clamped in-bounds, its stale data is
  // annihilated by B==0 (finite inputs, so no NaN/Inf propagation).
  const int h   = lane >> 4;        // which K-half this lane supplies
  const int rit = lane & 15;        // row-in-tile for the A operand
  const int rowA0 = (base + rit) * WIDE_NDENSE;        // tile 0: rows base+0..15
  const int rowA1 = (base + 16 + rit) * WIDE_NDENSE;   // tile 1: rows base+16..31
  v8f d0 = {0.f, 0.f, 0.f, 0.f, 0.f, 0.f, 0.f, 0.f};
  v8f d1 = {0.f, 0.f, 0.f, 0.f, 0.f, 0.f, 0.f, 0.f};
  #pragma unroll
  for (int kt = 0; kt < 4; ++kt) {
    const int k0 = 4 * kt + 2 * h;                     // this lane's even K slot
    v2f a0, a1, b;
    a0[0] = dense[min(rowA0 + k0,     WIDE_DENSE_LAST)];
    a0[1] = dense[min(rowA0 + k0 + 1, WIDE_DENSE_LAST)];
    a1[0] = dense[min(rowA1 + k0,     WIDE_DENSE_LAST)];
    a1[1] = dense[min(rowA1 + k0 + 1, WIDE_DENSE_LAST)];
    const float w0 = W[k0];                            // always in-bounds (k0 <= 15)
    const float w1 = W[k0 + 1];
    b[0] = (k0     < WIDE_NDENSE) ? w0 : 0.f;          // value select -> v_cndmask
    b[1] = (k0 + 1 < WIDE_NDENSE) ? w1 : 0.f;
    // 8 args: (neg_a, A, neg_b, B, c_mod, C, reuse_a, reuse_b)
    d0 = __builtin_amdgcn_wmma_f32_16x16x4_f32(false, a0, false, b, (short)0, d0,
                                               false, false);
    d1 = __builtin_amdgcn_wmma_f32_16x16x4_f32(false, a1, false, b, (short)0, d1,
                                               false, false);
  }
  // ---- redistribute: lane r wants row base+r ------------------------------
  // D layout: VGPR e, lanes 0-15 -> row e; lanes 16-31 -> row e+8 (cols identical).
  // Source lane s (j = s&15) publishes: j<8 -> d0[j], else d1[j-8].
  // Destination r reads from  s = ((r>>3)&1)<<4 | (r&16)>>1 | (r&7).
  const int j     = lane & 15;
  const float pub = (j < 8) ? sel8(d0, j) : sel8(d1, j - 8);
  const int src   = (((lane >> 3) & 1) << 4) | ((lane & 16) >> 1) | (lane & 7);
  dval = __shfl(pub, src, 32);
#else
  // scalar fallback: 13 FMAs per row
  float acc = 0.f;
  #pragma unroll
  for (int k = 0; k < WIDE_NDENSE; ++k)
    acc += dense[tid * WIDE_NDENSE + k] * W[k];
  dval = acc;
#endif

  out[tid] = sacc + dval;   // coalesced 32-float store per wave
}

extern "C" void kernel_launch(void* const* d_in, const int* in_sizes, int n_in,
                              void* d_out, int out_size, void* d_ws, size_t ws_size,
                              hipStream_t stream) {
  (void)in_sizes; (void)n_in; (void)out_size; (void)d_ws; (void)ws_size;
  const float* dense = (const float*)d_in[0];
  const float* W     = (const float*)d_in[1];
  const int* s0 = (const int*)d_in[2];
  const int* s1 = (const int*)d_in[3];
  const int* s2 = (const int*)d_in[4];
  const int* s3 = (const int*)d_in[5];
  const int* s4 = (const int*)d_in[6];
  const int* s5 = (const int*)d_in[7];
  const int* s6 = (const int*)d_in[8];
  const int* s7 = (const int*)d_in[9];
  float* out = (float*)d_out;

  dim3 grid(WIDE_BATCH / WIDE_BLOCK);   // 128 blocks x 4 waves -> spread over WGPs
  dim3 block(WIDE_BLOCK);
  wide_gather_wmma_kernel<<<grid, block, 0, stream>>>(dense, W,
                                                      s0, s1, s2, s3, s4, s5, s6, s7,
                                                      out);
}